// SPSE_5566277615910
// MI455X (gfx1250) — compile-verified
//
#include <hip/hip_runtime.h>
#include <hip/hip_bf16.h>

// Problem constants (from reference)
#define B_  16
#define C_  256
#define H_  128
#define W_  128
#define PLANE (H_ * W_)   // 16384 elements per (b,c)

typedef __attribute__((ext_vector_type(2))) float v2f;
typedef __attribute__((ext_vector_type(4))) float v4f;
typedef __attribute__((ext_vector_type(8))) float v8f;

// ---------------------------------------------------------------------------
// Kernel 1: multiscale max pool + per-channel 21->1 fusion linear.
// One block per (b,c) plane. 256 threads = 16 threads per 32x32 region.
// The rate-4 regions (4x4 grid of 32x32 tiles) nest exactly inside the
// rate-2 regions (2x2 grid of 64x64) and the rate-1 global max, so we only
// reduce the 16 region maxes once and derive the other 5 pooled values.
// ---------------------------------------------------------------------------
__global__ void pool_fuse_kernel(const float* __restrict__ x,
                                 const float* __restrict__ Wf,
                                 const float* __restrict__ bf,
                                 float* __restrict__ fused) {
    const int bc = blockIdx.x;          // b*C + c
    const int c  = bc & (C_ - 1);
    const float* plane = x + (size_t)bc * PLANE;

    const int t      = threadIdx.x;     // 0..255
    const int region = t >> 4;          // 0..15  (ry*4 + rx)
    const int sub    = t & 15;          // 0..15
    const int ry = region >> 2;
    const int rx = region & 3;
    const int row0 = ry * 32 + sub * 2; // two rows of 32 cols per thread
    const int col0 = rx * 32;

    const v4f* r0 = (const v4f*)(plane + row0 * W_ + col0);
    const v4f* r1 = (const v4f*)(plane + (row0 + 1) * W_ + col0);

    float m = -__builtin_inff();
#pragma unroll
    for (int i = 0; i < 8; ++i) {
        v4f a = r0[i];
        v4f b = r1[i];
        m = fmaxf(m, fmaxf(fmaxf(a.x, a.y), fmaxf(a.z, a.w)));
        m = fmaxf(m, fmaxf(fmaxf(b.x, b.y), fmaxf(b.z, b.w)));
    }

    // Reduce across the 16 lanes of this region (contiguous 16-lane groups
    // inside each wave32 -> width-16 xor shuffles stay in-group).
#pragma unroll
    for (int off = 8; off >= 1; off >>= 1)
        m = fmaxf(m, __shfl_xor(m, off, 16));

    __shared__ float r16s[16];
    if (sub == 0) r16s[region] = m;
    __syncthreads();

    if (t == 0) {
        float r16v[16];
        float r1max = -__builtin_inff();
#pragma unroll
        for (int i = 0; i < 16; ++i) {
            r16v[i] = r16s[i];
            r1max = fmaxf(r1max, r16v[i]);
        }
        // rate-2 maxes: 2x2 grid, each = max of 2x2 block of rate-4 regions
        float r2v[4];
#pragma unroll
        for (int j = 0; j < 4; ++j) {
            int jy = j >> 1, jx = j & 1;
            int base = (jy * 2) * 4 + (jx * 2);
            r2v[j] = fmaxf(fmaxf(r16v[base], r16v[base + 1]),
                           fmaxf(r16v[base + 4], r16v[base + 5]));
        }
        // fused[b,c] = Wf[c,:] . [r1 | r2(4) | r4(16)] + bf[c]
        const float* wrow = Wf + c * 21;
        float acc = bf[c] + wrow[0] * r1max;
#pragma unroll
        for (int j = 0; j < 4; ++j)  acc += wrow[1 + j] * r2v[j];
#pragma unroll
        for (int i = 0; i < 16; ++i) acc += wrow[5 + i] * r16v[i];
        fused[bc] = acc;
    }
}

// ---------------------------------------------------------------------------
// Kernel 2: SE bottleneck on tensor hardware. Single wave32.
//   h = relu(fused(16x256) @ W1(256x16) + b1)   -> 64x V_WMMA_F32_16X16X4_F32
//   s = h(16x16) @ W2(16x256) + b2              -> 16 tiles x 4 WMMAs
// A-fragment layout (16x4 f32 A, ISA 7.12.2): lane l<16 holds M=l,
// {K=k0,k0+1}; lane 16+l holds M=l, {K=k0+2,k0+3}.
// B-fragment: VGPR0 lanes0-15 = row K=k0 (N=0..15), lanes16-31 = row K=k0+2;
// VGPR1 = rows k0+1 / k0+3.
// D tile: VGPR v, lane n<16 -> (M=v, N=n); lane 16+n -> (M=8+v, N=n).
// ---------------------------------------------------------------------------
__global__ void se_wmma_kernel(const float* __restrict__ fused,
                               const float* __restrict__ W1,
                               const float* __restrict__ b1,
                               const float* __restrict__ W2,
                               const float* __restrict__ b2,
                               float* __restrict__ s) {
    const int lane = threadIdx.x;   // 0..31, full wave, EXEC all-1s
    const int half = lane >> 4;     // 0 or 1
    const int l    = lane & 15;

    // ---- h = fused @ W1 ----
    v8f acc = {};
    for (int k0 = 0; k0 < C_; k0 += 4) {
        const int ka = k0 + 2 * half;
        v2f a, b;
        a.x = fused[l * C_ + ka + 0];
        a.y = fused[l * C_ + ka + 1];
        b.x = W1[(ka + 0) * 16 + l];
        b.y = W1[(ka + 1) * 16 + l];
        acc = __builtin_amdgcn_wmma_f32_16x16x4_f32(
            /*neg_a=*/false, a, /*neg_b=*/false, b,
            /*c_mod=*/(short)0, acc, /*reuse_a=*/false, /*reuse_b=*/false);
    }

    // bias + relu; this lane owns column n=l, rows v + 8*half
    const float bias1 = b1[l];
    __shared__ float hbuf[16 * 16];
#pragma unroll
    for (int v = 0; v < 8; ++v) {
        float hv = fmaxf(acc[v] + bias1, 0.0f);
        hbuf[(v + 8 * half) * 16 + l] = hv;   // row-major h in LDS
    }
    __syncthreads();

    // ---- s = h @ W2 + b2, 16 N-tiles of 16 columns ----
    for (int nt = 0; nt < 16; ++nt) {
        v8f d = {};
#pragma unroll
        for (int k0 = 0; k0 < 16; k0 += 4) {
            const int ka = k0 + 2 * half;
            v2f a, b;
            a.x = hbuf[l * 16 + ka + 0];
            a.y = hbuf[l * 16 + ka + 1];
            b.x = W2[(ka + 0) * C_ + nt * 16 + l];
            b.y = W2[(ka + 1) * C_ + nt * 16 + l];
            d = __builtin_amdgcn_wmma_f32_16x16x4_f32(
                false, a, false, b, (short)0, d, false, false);
        }
        const int col   = nt * 16 + l;
        const float bc2 = b2[col];
#pragma unroll
        for (int v = 0; v < 8; ++v)
            s[(v + 8 * half) * C_ + col] = d[v] + bc2;
    }
}

// ---------------------------------------------------------------------------
// Kernel 3: out = x * s[b,c] broadcast. One block per (b,c) plane.
// float4 streaming loads; nontemporal stores (output is never re-read).
// ---------------------------------------------------------------------------
__global__ void scale_kernel(const float* __restrict__ x,
                             const float* __restrict__ s,
                             float* __restrict__ out) {
    const int bc = blockIdx.x;
    const float scale = s[bc];
    const v4f* xin = (const v4f*)(x + (size_t)bc * PLANE);
    v4f*       o   = (v4f*)(out + (size_t)bc * PLANE);
#pragma unroll 4
    for (int i = threadIdx.x; i < PLANE / 4; i += 256) {
        v4f v = xin[i];
        v *= scale;
        __builtin_nontemporal_store(v, &o[i]);
    }
}

extern "C" void kernel_launch(void* const* d_in, const int* in_sizes, int n_in,
                              void* d_out, int out_size, void* d_ws, size_t ws_size,
                              hipStream_t stream) {
    const float* x  = (const float*)d_in[0];   // (B,C,H,W)
    const float* Wf = (const float*)d_in[1];   // (C,21)
    const float* bf = (const float*)d_in[2];   // (C,)
    const float* W1 = (const float*)d_in[3];   // (C,16)
    const float* b1 = (const float*)d_in[4];   // (16,)
    const float* W2 = (const float*)d_in[5];   // (16,C)
    const float* b2 = (const float*)d_in[6];   // (C,)
    float* out = (float*)d_out;

    float* fused = (float*)d_ws;               // B*C floats
    float* svec  = fused + B_ * C_;            // B*C floats

    pool_fuse_kernel<<<B_ * C_, 256, 0, stream>>>(x, Wf, bf, fused);
    se_wmma_kernel<<<1, 32, 0, stream>>>(fused, W1, b1, W2, b2, svec);
    scale_kernel<<<B_ * C_, 256, 0, stream>>>(x, svec, out);
}